// Izhikevich_24575802868466
// MI455X (gfx1250) — compile-verified
//
#include <hip/hip_runtime.h>
#include <math.h>

// Izhikevich network step for MI455X (gfx1250, wave32).
//
// Cost model: I = w @ g is an 8192x8192 fp32 GEMV -> 256 MB of w streamed
// once. At 23.3 TB/s the floor is ~11 us; FLOPs (134 MFLOP) are free.
// We run the reduction through V_WMMA_F32_16X16X4_F32 so the accumulate
// chain sits in the matrix pipe:
//   A tile  = 16 rows x 4 cols of w   (per-lane float2 at col k0+2*(lane/16))
//   B tile  = g[k..k+3] replicated over all 16 N-columns (same float2 shape)
//   C tile  = every column identical = the 16 partial dot products.
// 512 blocks x 8 wave32 (split-K over 8 slices of 1024) = 4096 waves to keep
// HBM saturated. Partials reduced via LDS; the elementwise Izhikevich tail is
// fused into the same kernel (threads 0..15 per block).

typedef float v2f __attribute__((ext_vector_type(2)));
typedef float v8f __attribute__((ext_vector_type(8)));

#define NN 8192
#define WAVES 8
#define KSLICE (NN / WAVES) /* 1024 columns per wave */

__global__ __launch_bounds__(256) void izhikevich_step(
    const float* __restrict__ x_in, const float* __restrict__ v_in,
    const float* __restrict__ u_in, const float* __restrict__ g_in,
    const float* __restrict__ w,    const float* __restrict__ a_p,
    const float* __restrict__ b_p,  const float* __restrict__ c_p,
    const float* __restrict__ d_p,  float* __restrict__ out)
{
    __shared__ float red[WAVES * 256]; // 8 KB: per-wave C tiles
    __shared__ float Ivals[16];        // combined dot products for this row tile

    const int tid  = threadIdx.x;
    const int wave = tid >> 5;
    const int lane = tid & 31;

    const int rowBase = blockIdx.x * 16;
    const int row     = rowBase + (lane & 15);
    const int kOff    = (lane >> 4) << 1;   // 0 for lanes 0-15, 2 for 16-31
    const int k0      = wave * KSLICE;

    // ISA layout, 32-bit A 16x4: lane l, VGPR v holds A[M=l%16][K=v+2*(l/16)]
    //             32-bit B 4x16: lane l, VGPR v holds B[K=v+2*(l/16)][N=l%16]
    // -> both operands are a contiguous float2 at column (k + kOff).
    const float* wrow = w    + (size_t)row * NN + (size_t)(k0 + kOff);
    const float* gptr = g_in + (k0 + kOff);

    v8f c = {};
#pragma unroll 8
    for (int k = 0; k < KSLICE; k += 4) {
        v2f a = *(const v2f*)(wrow + k);   // 8B-aligned: offsets always even
        v2f b = *(const v2f*)(gptr + k);
        // (neg_a, A, neg_b, B, c_mod, C, reuse_a, reuse_b)
        c = __builtin_amdgcn_wmma_f32_16x16x4_f32(
                false, a, false, b, (short)0, c, false, false);
    }

    // ---- split-K reduction across the 8 waves via LDS ----
#pragma unroll
    for (int r = 0; r < 8; ++r)
        red[wave * 256 + lane * 8 + r] = c[r];
    __syncthreads();

    {
        // tid indexes (lane = tid/8, vgpr r = tid%8) of the C layout.
        float s = 0.0f;
#pragma unroll
        for (int wv = 0; wv < WAVES; ++wv)
            s += red[wv * 256 + tid];
        const int l = tid >> 3;
        const int r = tid & 7;
        const int n = l & 15;                  // C column
        const int m = r + ((l >> 4) << 3);     // C row
        if (n == 0)                            // all columns identical; take N=0
            Ivals[m] = s;
    }
    __syncthreads();

    // ---- fused elementwise Izhikevich update for this block's 16 rows ----
    if (tid < 16) {
        const int i = rowBase + tid;
        const float I  = Ivals[tid] + x_in[i];
        const float v  = v_in[i];
        const float u  = u_in[i];
        const float g  = g_in[i];

        const float v1 = v + (0.04f * v * v + 5.0f * v + 140.0f - u + I);

        // differentiable spike surrogate
        const float ss = 1.0f / (1.0f + __expf(-4.0f * (v1 - 30.0f)));
        // hard spike mask for resets
        const float sp = (v1 >= 30.0f) ? 1.0f : 0.0f;
        const float ns = 1.0f - sp;

        const float dg = -g / 6.5f;
        const float du = fabsf(a_p[i]) * (fabsf(b_p[i]) * v1 - u);

        out[i]           = ns * v1 + sp * c_p[i];        // v_out
        out[NN + i]      = ss;                           // spiked_s
        out[2 * NN + i]  = ns * (u + du) + sp * d_p[i];  // u_out
        out[3 * NN + i]  = ns * (g + dg) + sp;           // g_out
    }
}

extern "C" void kernel_launch(void* const* d_in, const int* in_sizes, int n_in,
                              void* d_out, int out_size, void* d_ws, size_t ws_size,
                              hipStream_t stream) {
    (void)in_sizes; (void)n_in; (void)out_size; (void)d_ws; (void)ws_size;

    const float* x_in = (const float*)d_in[0];
    const float* v    = (const float*)d_in[1];
    const float* u    = (const float*)d_in[2];
    const float* g    = (const float*)d_in[3];
    const float* w    = (const float*)d_in[4];
    const float* a_p  = (const float*)d_in[5];
    const float* b_p  = (const float*)d_in[6];
    const float* c_p  = (const float*)d_in[7];
    const float* d_p  = (const float*)d_in[8];
    float* out = (float*)d_out;

    dim3 grid(NN / 16);   // 512 row tiles
    dim3 block(256);      // 8 wave32 per block, split-K over 8 slices
    izhikevich_step<<<grid, block, 0, stream>>>(
        x_in, v, u, g, w, a_p, b_p, c_p, d_p, out);
}